// PointNetMSG_73718818669267
// MI455X (gfx1250) — compile-verified
//
#include <hip/hip_runtime.h>
#include <hip/hip_bf16.h>

// ---------------- problem constants ----------------
#define B_  4
#define N_  16384
#define C_  32
#define M_  1024
#define K0_ 32
#define K1_ 64
#define R0SQ 0.01f
#define R1SQ 0.04f

typedef float v2f __attribute__((ext_vector_type(2)));
typedef float v8f __attribute__((ext_vector_type(8)));
typedef unsigned int v4u __attribute__((ext_vector_type(4)));
typedef int v8i __attribute__((ext_vector_type(8)));
typedef int v4i __attribute__((ext_vector_type(4)));

// ============================================================
// Kernel 1: materialize support = concat(xyz, feats) and |x|^2
// ============================================================
__global__ void build_support_kernel(const float* __restrict__ xyz,
                                     const float* __restrict__ pf,
                                     float* __restrict__ sup,
                                     float* __restrict__ xn) {
    int i = blockIdx.x * blockDim.x + threadIdx.x;      // B*N threads
    if (i >= B_ * N_) return;
    int b = i >> 14, n = i & (N_ - 1);
    float acc = 0.f;
#pragma unroll
    for (int c = 0; c < C_; ++c) {
        float v = (c < 3) ? xyz[((size_t)b * 3 + c) * N_ + n]
                          : pf[((size_t)b * 29 + (c - 3)) * N_ + n];
        sup[((size_t)b * C_ + c) * N_ + n] = v;
        acc += v * v;
    }
    xn[(size_t)b * N_ + n] = acc;
}

// ============================================================
// Kernel 2: farthest point sampling (1 block per batch)
// ============================================================
#define FPS_T 1024
#define FPS_P (N_ / FPS_T)   // 16 points per thread

__global__ __launch_bounds__(FPS_T) void fps_kernel(const float* __restrict__ sup,
                                                    int* __restrict__ cidx) {
    const int b = blockIdx.x;
    const int t = threadIdx.x;
    const int lane = t & 31, wid = t >> 5;   // 32 waves
    __shared__ float sCent[C_];
    __shared__ float sBest[32];
    __shared__ int   sBidx[32];
    __shared__ int   sFar;

    const float* supb = sup + (size_t)b * C_ * N_;
    float dist[FPS_P];
#pragma unroll
    for (int i = 0; i < FPS_P; ++i) dist[i] = 1e10f;
    int far = 0;

    for (int it = 0; it < M_; ++it) {
        if (t == 0) cidx[b * M_ + it] = far;
        if (t < C_) sCent[t] = supb[t * N_ + far];
        __syncthreads();
        float cent[C_];
#pragma unroll
        for (int c = 0; c < C_; ++c) cent[c] = sCent[c];

        float best = -1.f; int bidx = 0;
#pragma unroll
        for (int i = 0; i < FPS_P; ++i) {
            int n = t + i * FPS_T;
            float acc = 0.f;
#pragma unroll
            for (int c = 0; c < C_; ++c) {
                float d = supb[c * N_ + n] - cent[c];
                acc = fmaf(d, d, acc);
            }
            float dm = fminf(dist[i], acc);
            dist[i] = dm;
            if (dm > best) { best = dm; bidx = n; }   // ascending n -> first occurrence
        }
        // wave reduce (argmax, prefer smaller index on tie)
#pragma unroll
        for (int off = 16; off > 0; off >>= 1) {
            float ob = __shfl_xor(best, off, 32);
            int   oi = __shfl_xor(bidx, off, 32);
            if (ob > best || (ob == best && oi < bidx)) { best = ob; bidx = oi; }
        }
        if (lane == 0) { sBest[wid] = best; sBidx[wid] = bidx; }
        __syncthreads();
        if (wid == 0) {
            best = sBest[lane]; bidx = sBidx[lane];
#pragma unroll
            for (int off = 16; off > 0; off >>= 1) {
                float ob = __shfl_xor(best, off, 32);
                int   oi = __shfl_xor(bidx, off, 32);
                if (ob > best || (ob == best && oi < bidx)) { best = ob; bidx = oi; }
            }
            if (lane == 0) sFar = bidx;
        }
        __syncthreads();
        far = sFar;
    }
}

// ============================================================
// Kernel 3: gather centroids, |c|^2, and new_xyz (-> d_out)
// ============================================================
__global__ void gather_centroids_kernel(const float* __restrict__ sup,
                                        const float* __restrict__ xyz,
                                        const int* __restrict__ cidx,
                                        float* __restrict__ cent,
                                        float* __restrict__ cn,
                                        float* __restrict__ out_newxyz) {
    int i = blockIdx.x * blockDim.x + threadIdx.x;    // B*M
    if (i >= B_ * M_) return;
    int b = i >> 10, m = i & (M_ - 1);
    int src = cidx[i];
    float acc = 0.f;
#pragma unroll
    for (int c = 0; c < C_; ++c) {
        float v = sup[((size_t)b * C_ + c) * N_ + src];
        cent[((size_t)b * C_ + c) * M_ + m] = v;
        acc += v * v;
    }
    cn[i] = acc;
#pragma unroll
    for (int c = 0; c < 3; ++c)
        out_newxyz[((size_t)b * 3 + c) * M_ + m] = xyz[((size_t)b * 3 + c) * N_ + src];
}

// ============================================================
// Kernel 4: fused dual-radius ball query
//   - support tile (32 x 256, stride-257 padded) fetched by the
//     Tensor Data Mover (tensor_load_to_lds + s_wait_tensorcnt)
//   - 16 waves each compute a 16x16 distance sub-tile with a
//     chain of 8 V_WMMA_F32_16X16X4_F32 (B-frags preloaded)
//   - in-order dual-radius selection via wave32 ballot/popcount
// ============================================================
#define BQ_T 512
#define TN   256

__global__ __launch_bounds__(BQ_T, 1) void ballquery_kernel(
        const float* __restrict__ sup, const float* __restrict__ xn,
        const float* __restrict__ cent, const float* __restrict__ cn,
        int* __restrict__ idx0, int* __restrict__ msk0,
        int* __restrict__ idx1, int* __restrict__ msk1) {
    const int blk = blockIdx.x;                 // B * (M/16) = 256
    const int b = blk >> 6;
    const int mtile = (blk & 63) * 16;
    const int t = threadIdx.x, lane = t & 31, wid = t >> 5;

    __shared__ float sC[16][C_ + 1];
    __shared__ float sX[C_][TN + 1];            // TDM pad gives stride TN+1
    __shared__ float sD[16][TN + 1];
    __shared__ float sXN[TN];
    __shared__ float sCN[16];
    __shared__ int   sIdx0[16][K0_];
    __shared__ int   sIdx1[16][K1_];
    __shared__ int   sRowFull[16];
    __shared__ int   sAllFull;

    for (int i = t; i < 16 * C_; i += BQ_T) {
        int r = i & 15, c = i >> 4;
        sC[r][c] = cent[((size_t)b * C_ + c) * M_ + mtile + r];
    }
    if (t < 16) { sCN[t] = cn[b * M_ + mtile + t]; sRowFull[t] = 0; }
    if (t == 0) sAllFull = 0;
    __syncthreads();

    // A fragments (16x32 centroid tile), persist over all n-tiles
    v2f afrag[8];
    {
        int row = lane & 15;
        int koff = (lane < 16) ? 0 : 2;
#pragma unroll
        for (int kc = 0; kc < 8; ++kc) {
            afrag[kc].x = sC[row][kc * 4 + koff + 0];
            afrag[kc].y = sC[row][kc * 4 + koff + 1];
        }
    }

#if __has_builtin(__builtin_amdgcn_tensor_load_to_lds)
    // TDM descriptor constants (per ISA D# layout).  2D tile:
    //   dim0 = points (256 contiguous), dim1 = channels (32, stride N)
    //   pad: +1 DWORD after every 256 DWORDs -> LDS row stride 257
    const unsigned ldsBase = (unsigned)(size_t)(&sX[0][0]);
    const unsigned g1w0 = (2u << 16)      // data_size = 4 bytes
                        | (1u << 20)      // pad_enable
                        | (7u << 22)      // pad_interval: 256 DWORDs
                        | (0u << 25);     // pad_amount: 1 DWORD
#endif

    int cnt0 = 0, cnt1 = 0;
    const int myrow = wid;
    const unsigned lmask_lt = (1u << lane) - 1u;

    for (int nt = 0; nt < N_ / TN; ++nt) {
        const int nbase = nt * TN;

#if __has_builtin(__builtin_amdgcn_tensor_load_to_lds)
        if (wid == 0) {   // one wave drives the DMA (EXEC-independent op)
            unsigned long long ga =
                (unsigned long long)(size_t)(sup + (size_t)b * C_ * N_ + nbase);
            v4u g0;
            g0[0] = 1u;                                   // count=1, user mode
            g0[1] = ldsBase;                              // lds_addr
            g0[2] = (unsigned)ga;                         // global_addr[31:0]
            g0[3] = (unsigned)(ga >> 32) | (2u << 30);    // addr[56:32] | type=2
            v8i g1;
            g1[0] = (int)g1w0;
            g1[1] = (int)((N_ & 0xFFFFu) << 16);          // tensor_dim0 lo16
            g1[2] = (int)((N_ >> 16) | (C_ << 16));       // dim0 hi16 | dim1 lo16
            g1[3] = (int)(TN << 16);                      // dim1 hi16 | tile_dim0
            g1[4] = (int)C_;                              // tile_dim1 | tile_dim2=0
            g1[5] = (int)N_;                              // dim0_stride lo32
            g1[6] = 0;                                    // stride hi | dim1_stride lo
            g1[7] = 0;
            v4i gz4 = {};
            v8i gz8 = {};
            __builtin_amdgcn_tensor_load_to_lds(g0, g1, gz4, gz4, gz8, 0);
            __builtin_amdgcn_s_wait_tensorcnt(0);
        }
#else
        for (int i = t; i < C_ * TN; i += BQ_T) {
            int c = i >> 8, x = i & (TN - 1);
            sX[c][x] = sup[((size_t)b * C_ + c) * N_ + nbase + x];
        }
#endif
        if (t < TN) sXN[t] = xn[(size_t)b * N_ + nbase + t];
        __syncthreads();

        // ---- phase A: wave `wid` computes 16x16 sub-tile (cols wid*16..) ----
        {
            const int c0 = wid * 16;
            const int col = lane & 15;
            const int kb = (lane < 16) ? 0 : 2;
            // preload all B fragments, then run the WMMA chain back-to-back
            v2f bfrag[8];
#pragma unroll
            for (int kc = 0; kc < 8; ++kc) {
                bfrag[kc].x = sX[kc * 4 + kb + 0][c0 + col];
                bfrag[kc].y = sX[kc * 4 + kb + 1][c0 + col];
            }
            v8f acc = {};
#pragma unroll
            for (int kc = 0; kc < 8; ++kc) {
#if __has_builtin(__builtin_amdgcn_wmma_f32_16x16x4_f32)
                acc = __builtin_amdgcn_wmma_f32_16x16x4_f32(
                        false, afrag[kc], false, bfrag[kc], (short)0, acc, false, false);
#else
                for (int r = 0; r < 8; ++r) {
                    int row = ((lane < 16) ? 0 : 8) + r;
                    acc[r] += sC[row][kc * 4 + 0] * sX[kc * 4 + 0][c0 + col]
                            + sC[row][kc * 4 + 1] * sX[kc * 4 + 1][c0 + col]
                            + sC[row][kc * 4 + 2] * sX[kc * 4 + 2][c0 + col]
                            + sC[row][kc * 4 + 3] * sX[kc * 4 + 3][c0 + col];
                }
#endif
            }
            const float cxn = sXN[c0 + col];
            const int rbase = (lane < 16) ? 0 : 8;
#pragma unroll
            for (int r = 0; r < 8; ++r) {
                int row = rbase + r;
                sD[row][c0 + col] = sCN[row] + cxn - 2.0f * acc[r];
            }
        }
        __syncthreads();

        // ---- phase B: wave `myrow` scans its centroid row, in n order ----
        if (!(cnt0 >= K0_ && cnt1 >= K1_)) {
#pragma unroll
            for (int ch = 0; ch < TN / 32; ++ch) {
                int col = ch * 32 + lane;
                float d = sD[myrow][col];
                bool p0 = d <= R0SQ;
                bool p1 = d <= R1SQ;
                unsigned m0 = (unsigned)__builtin_amdgcn_ballot_w32(p0);
                unsigned m1 = (unsigned)__builtin_amdgcn_ballot_w32(p1);
                int pos0 = cnt0 + __builtin_popcount(m0 & lmask_lt);
                int pos1 = cnt1 + __builtin_popcount(m1 & lmask_lt);
                int n = nbase + col;
                if (p0 && pos0 < K0_) sIdx0[myrow][pos0] = n;
                if (p1 && pos1 < K1_) sIdx1[myrow][pos1] = n;
                cnt0 += __builtin_popcount(m0);
                cnt1 += __builtin_popcount(m1);
            }
        }
        if (lane == 0) sRowFull[myrow] = (cnt0 >= K0_ && cnt1 >= K1_) ? 1 : 0;
        __syncthreads();
        if (t == 0) {
            int all = 1;
            for (int r = 0; r < 16; ++r) all &= sRowFull[r];
            sAllFull = all;
        }
        __syncthreads();
        if (sAllFull) break;
    }

    // ---- epilogue: pad-with-first + mask, write both scales ----
    {
        const int m = mtile + myrow;
        const size_t base0 = ((size_t)b * M_ + m) * K0_;
        const size_t base1 = ((size_t)b * M_ + m) * K1_;
        int v0 = cnt0 < K0_ ? cnt0 : K0_;
        int v1 = cnt1 < K1_ ? cnt1 : K1_;
        int first0 = (v0 > 0) ? sIdx0[myrow][0] : 0;
        int first1 = (v1 > 0) ? sIdx1[myrow][0] : 0;
        if (lane < K0_) {
            idx0[base0 + lane] = (lane < v0) ? sIdx0[myrow][lane] : first0;
            msk0[base0 + lane] = (lane < v0) ? 0 : 1;
        }
        for (int j = lane; j < K1_; j += 32) {
            idx1[base1 + j] = (j < v1) ? sIdx1[myrow][j] : first1;
            msk1[base1 + j] = (j < v1) ? 0 : 1;
        }
    }
}

// ============================================================
// Kernel 5: per-centroid attention (algebraically collapsed):
//   w_h = (Wk_h^T q_h) . y ;  o_h = Wv_h (sum_j softmax(w)_j y_j)
// ============================================================
__global__ void attention_kernel(const float* __restrict__ sup,
                                 const float* __restrict__ cent,
                                 const int* __restrict__ idx,
                                 const int* __restrict__ msk, int K,
                                 const float* __restrict__ wq,
                                 const float* __restrict__ wk,
                                 const float* __restrict__ wv,
                                 const float* __restrict__ wo,
                                 float* __restrict__ fatt) {
    int i = blockIdx.x * blockDim.x + threadIdx.x;      // B*M
    if (i >= B_ * M_) return;
    int b = i >> 10, m = i & (M_ - 1);

    float cvec[C_];
#pragma unroll
    for (int c = 0; c < C_; ++c) cvec[c] = cent[((size_t)b * C_ + c) * M_ + m];

    float qk[2][C_];
    {
        float q[64];
#pragma unroll 4
        for (int o = 0; o < 64; ++o) {
            float a = 0.f;
#pragma unroll
            for (int c = 0; c < C_; ++c) a = fmaf(wq[o * C_ + c], cvec[c], a);
            q[o] = a;
        }
        for (int h = 0; h < 2; ++h)
#pragma unroll 4
            for (int cp = 0; cp < C_; ++cp) {
                float a = 0.f;
#pragma unroll
                for (int c = 0; c < C_; ++c)
                    a = fmaf(q[h * 32 + c], wk[(h * 32 + c) * C_ + cp], a);
                qk[h][cp] = a;
            }
    }

    const float inv_sqrtC = 0.17677669529663687f;  // 1/sqrt(32)
    float mrun[2] = {-3.0e38f, -3.0e38f};
    float den[2]  = {0.f, 0.f};
    float ysum[2][C_];
    for (int h = 0; h < 2; ++h)
#pragma unroll
        for (int c = 0; c < C_; ++c) ysum[h][c] = 0.f;

    const size_t base = ((size_t)b * M_ + m) * K;
    for (int j = 0; j < K; ++j) {
        int n  = idx[base + j];
        int mk = msk[base + j];
        float y[C_];
#pragma unroll
        for (int c = 0; c < C_; ++c)
            y[c] = sup[((size_t)b * C_ + c) * N_ + n] - cvec[c];
        for (int h = 0; h < 2; ++h) {
            float l;
            if (mk) l = -1e9f;
            else {
                l = 0.f;
#pragma unroll
                for (int c = 0; c < C_; ++c) l = fmaf(qk[h][c], y[c], l);
                l *= inv_sqrtC;
            }
            float mn = fmaxf(mrun[h], l);
            float sc = __expf(mrun[h] - mn);
            float e  = __expf(l - mn);
            den[h] = den[h] * sc + e;
#pragma unroll
            for (int c = 0; c < C_; ++c) ysum[h][c] = ysum[h][c] * sc + e * y[c];
            mrun[h] = mn;
        }
    }

    float ovec[64];
    for (int h = 0; h < 2; ++h) {
        float invd = 1.0f / den[h];
#pragma unroll 4
        for (int oc = 0; oc < 32; ++oc) {
            float a = 0.f;
#pragma unroll
            for (int cp = 0; cp < C_; ++cp)
                a = fmaf(wv[(h * 32 + oc) * C_ + cp], ysum[h][cp], a);
            ovec[h * 32 + oc] = a * invd;
        }
    }
#pragma unroll 4
    for (int c = 0; c < C_; ++c) {
        float a = cvec[c];
#pragma unroll
        for (int o = 0; o < 64; ++o) a = fmaf(wo[c * 64 + o], ovec[o], a);
        fatt[((size_t)b * C_ + c) * M_ + m] = a;
    }
}

// ============================================================
// Kernels 6-8: 1x1 conv (GEMM) + bias, BN stats, BN normalize
// ============================================================
__global__ void gemm_bias_kernel(const float* __restrict__ W,
                                 const float* __restrict__ bias,
                                 const float* __restrict__ in,
                                 float* __restrict__ out, int Cin, int Cout) {
    int i = blockIdx.x * blockDim.x + threadIdx.x;
    if (i >= B_ * Cout * M_) return;
    int m = i & (M_ - 1);
    int o = (i >> 10) % Cout;
    int b = i / (Cout * M_);
    float a = bias[o];
    for (int c = 0; c < Cin; ++c)
        a = fmaf(W[o * Cin + c], in[((size_t)b * Cin + c) * M_ + m], a);
    out[((size_t)b * Cout + o) * M_ + m] = a;
}

__global__ void bn_stats_kernel(const float* __restrict__ x, int Cout,
                                float* __restrict__ mu, float* __restrict__ var) {
    int o = blockIdx.x;       // one block per channel
    int t = threadIdx.x;
    __shared__ float ss[256], sq[256];
    float s = 0.f, q = 0.f;
    for (int i = t; i < B_ * M_; i += 256) {
        int b = i >> 10, m = i & (M_ - 1);
        float v = x[((size_t)b * Cout + o) * M_ + m];
        s += v; q += v * v;
    }
    ss[t] = s; sq[t] = q;
    __syncthreads();
    for (int off = 128; off > 0; off >>= 1) {
        if (t < off) { ss[t] += ss[t + off]; sq[t] += sq[t + off]; }
        __syncthreads();
    }
    if (t == 0) {
        float mu_ = ss[0] * (1.0f / (B_ * M_));
        mu[o] = mu_;
        var[o] = sq[0] * (1.0f / (B_ * M_)) - mu_ * mu_;
    }
}

__global__ void bn_norm_kernel(const float* __restrict__ x,
                               const float* __restrict__ mu,
                               const float* __restrict__ var,
                               const float* __restrict__ g,
                               const float* __restrict__ beta,
                               int Cout, int act,
                               float* __restrict__ dst, int dstC, int choff) {
    int i = blockIdx.x * blockDim.x + threadIdx.x;
    if (i >= B_ * Cout * M_) return;
    int m = i & (M_ - 1);
    int o = (i >> 10) % Cout;
    int b = i / (Cout * M_);
    float v = x[((size_t)b * Cout + o) * M_ + m];
    v = (v - mu[o]) * rsqrtf(var[o] + 1e-5f) * g[o] + beta[o];
    if (act) v = (v > 0.f) ? v : 0.02f * v;
    dst[((size_t)b * dstC + choff + o) * M_ + m] = v;
}

// ============================================================
// host-side launcher
// ============================================================
extern "C" void kernel_launch(void* const* d_in, const int* in_sizes, int n_in,
                              void* d_out, int out_size, void* d_ws, size_t ws_size,
                              hipStream_t stream) {
    const float* xyz = (const float*)d_in[0];
    const float* pf  = (const float*)d_in[1];

    // workspace carve-up (floats)
    float* ws   = (float*)d_ws;
    float* sup  = ws;                       size_t off = (size_t)B_ * C_ * N_;
    float* xn   = ws + off;                 off += (size_t)B_ * N_;
    int*   cidx = (int*)(ws + off);         off += (size_t)B_ * M_;
    float* cen  = ws + off;                 off += (size_t)B_ * C_ * M_;
    float* cn   = ws + off;                 off += (size_t)B_ * M_;
    int*   idx0 = (int*)(ws + off);         off += (size_t)B_ * M_ * K0_;
    int*   msk0 = (int*)(ws + off);         off += (size_t)B_ * M_ * K0_;
    int*   idx1 = (int*)(ws + off);         off += (size_t)B_ * M_ * K1_;
    int*   msk1 = (int*)(ws + off);         off += (size_t)B_ * M_ * K1_;
    float* fatt = ws + off;                 off += (size_t)B_ * C_ * M_;
    float* h1   = ws + off;                 off += (size_t)B_ * 64 * M_;
    float* h2   = ws + off;                 off += (size_t)B_ * 128 * M_;
    float* mu   = ws + off;                 off += 128;
    float* var  = ws + off;                 off += 128;

    float* out_newxyz = (float*)d_out;                       // [B,3,M]
    float* out_feats  = (float*)d_out + (size_t)B_ * 3 * M_; // [B,256,M]

    build_support_kernel<<<(B_ * N_) / 256, 256, 0, stream>>>(xyz, pf, sup, xn);
    fps_kernel<<<B_, FPS_T, 0, stream>>>(sup, cidx);
    gather_centroids_kernel<<<(B_ * M_) / 256, 256, 0, stream>>>(
        sup, xyz, cidx, cen, cn, out_newxyz);
    ballquery_kernel<<<B_ * (M_ / 16), BQ_T, 0, stream>>>(
        sup, xn, cen, cn, idx0, msk0, idx1, msk1);

    for (int s = 0; s < 2; ++s) {
        const int base = 2 + s * 12;
        const float* aq  = (const float*)d_in[base + 0];
        const float* ak  = (const float*)d_in[base + 1];
        const float* av  = (const float*)d_in[base + 2];
        const float* ao  = (const float*)d_in[base + 3];
        const float* mw0 = (const float*)d_in[base + 4];
        const float* mb0 = (const float*)d_in[base + 5];
        const float* mg0 = (const float*)d_in[base + 6];
        const float* mt0 = (const float*)d_in[base + 7];
        const float* mw1 = (const float*)d_in[base + 8];
        const float* mb1 = (const float*)d_in[base + 9];
        const float* mg1 = (const float*)d_in[base + 10];
        const float* mt1 = (const float*)d_in[base + 11];

        const int* idxS = (s == 0) ? idx0 : idx1;
        const int* mskS = (s == 0) ? msk0 : msk1;
        const int  K    = (s == 0) ? K0_ : K1_;

        attention_kernel<<<(B_ * M_) / 256, 256, 0, stream>>>(
            sup, cen, idxS, mskS, K, aq, ak, av, ao, fatt);

        gemm_bias_kernel<<<(B_ * 64 * M_) / 256, 256, 0, stream>>>(
            mw0, mb0, fatt, h1, 32, 64);
        bn_stats_kernel<<<64, 256, 0, stream>>>(h1, 64, mu, var);
        bn_norm_kernel<<<(B_ * 64 * M_) / 256, 256, 0, stream>>>(
            h1, mu, var, mg0, mt0, 64, 1, h1, 64, 0);

        gemm_bias_kernel<<<(B_ * 128 * M_) / 256, 256, 0, stream>>>(
            mw1, mb1, h1, h2, 64, 128);
        bn_stats_kernel<<<128, 256, 0, stream>>>(h2, 128, mu, var);
        bn_norm_kernel<<<(B_ * 128 * M_) / 256, 256, 0, stream>>>(
            h2, mu, var, mg1, mt1, 128, 0, out_feats, 256, s * 128);
    }
    (void)in_sizes; (void)n_in; (void)out_size; (void)ws_size;
}